// ConditionalRandomField_83863531422183
// MI455X (gfx1250) — compile-verified
//
#include <hip/hip_runtime.h>

typedef float v2f __attribute__((ext_vector_type(2)));
typedef float v8f __attribute__((ext_vector_type(8)));

#define B_TOT 512
#define T_LEN 1024
#define K_ST  64
#define TK    (T_LEN * K_ST)
#define WAVES_PER_BLOCK 4
#define ROWS_PER_WAVE   16
#define PROW 68   // padded LDS row stride (dwords) -> conflict-free A-fragment loads

// ---------------------------------------------------------------------------
// Forward (log-partition) kernel.
// One wave owns 16 batch rows x 64 states for the entire T recursion.
// score kept in WMMA C-layout: acc[ct][i], lane -> row = (lane>=16?8:0)+i,
// col = 16*ct + (lane&15).  Per step:
//   m   = rowmax(score)                      (VALU max + shfl_xor butterfly)
//   p   = exp(score - m)    -> LDS (row-major, pad 68) -> reload as A-fragments
//   r   = p @ E             (16 chained V_WMMA_F32_16X16X4_F32 per 16x16 tile)
//   sc' = em[t] + m + log(r), masked select
// ---------------------------------------------------------------------------
__global__ __launch_bounds__(128) void crf_forward_kernel(
    const float* __restrict__ em,            // (B,T,K)
    const unsigned char* __restrict__ mask,  // (B,T)
    const float* __restrict__ trans,         // (K,K)
    const float* __restrict__ startT,        // (K)
    const float* __restrict__ endT,          // (K)
    float* __restrict__ logZ)                // (B) in workspace
{
    // E in B-fragment layout: float2 per (ks, ct, lane); lane-stride-8B loads
    __shared__ float eFrag[16 * 4 * 32 * 2];                  // 16 KB
    __shared__ float pBuf[WAVES_PER_BLOCK][ROWS_PER_WAVE * PROW]; // 17 KB
    __shared__ float sStart[K_ST];
    __shared__ float sEnd[K_ST];

    const int tid  = threadIdx.x;
    const int lane = tid & 31;
    const int wave = tid >> 5;
    const int li   = lane & 15;
    const int hi   = lane >> 4;        // 0 or 1 (half-wave)
    const int rbase = hi << 3;         // 0 or 8

    // ---- build E = exp(trans) directly in B-fragment layout ----
    // B 4x16 tile, 2 VGPRs: v0: lanes0-15 K=0, lanes16-31 K=2; v1: K=1 / K=3
    for (int idx = tid; idx < 16 * 4 * 32; idx += blockDim.x) {
        int l  = idx & 31;
        int ct = (idx >> 5) & 3;
        int ks = idx >> 7;
        int col = 16 * ct + (l & 15);
        int r0  = 4 * ks + ((l >= 16) ? 2 : 0);
        eFrag[idx * 2 + 0] = __expf(trans[(r0 + 0) * K_ST + col]);
        eFrag[idx * 2 + 1] = __expf(trans[(r0 + 1) * K_ST + col]);
    }
    for (int idx = tid; idx < K_ST; idx += blockDim.x) {
        sStart[idx] = startT[idx];
        sEnd[idx]   = endT[idx];
    }
    __syncthreads();

    const int b0 = (blockIdx.x * WAVES_PER_BLOCK + wave) * ROWS_PER_WAVE;
    float* pw = pBuf[wave];

    // ---- t = 0: score = start + em[:,0,:]  (built in C-layout) ----
    v8f sc[4];
    #pragma unroll
    for (int ct = 0; ct < 4; ++ct) {
        int col = ct * 16 + li;
        float st = sStart[col];
        #pragma unroll
        for (int i = 0; i < 8; ++i) {
            long long boff = (long long)(b0 + rbase + i) * TK;
            sc[ct][i] = st + em[boff + col];
        }
    }

    // ---- recursion over T ----
    for (int t = 1; t < T_LEN; ++t) {
        // prefetch em tile for t+8 (2 cachelines per row, 32 lanes cover 16x64)
        if (t + 8 < T_LEN) {
            const float* pf = &em[(long long)(b0 + (lane >> 1)) * TK
                                  + (long long)(t + 8) * K_ST + (lane & 1) * 32];
            __builtin_prefetch(pf, 0, 1);
        }

        // row max over all 64 columns
        v8f m = sc[0];
        #pragma unroll
        for (int ct = 1; ct < 4; ++ct)
            #pragma unroll
            for (int i = 0; i < 8; ++i) m[i] = fmaxf(m[i], sc[ct][i]);
        #pragma unroll
        for (int off = 8; off >= 1; off >>= 1)
            #pragma unroll
            for (int i = 0; i < 8; ++i)
                m[i] = fmaxf(m[i], __shfl_xor(m[i], off));

        // p = exp(score - m) staged to LDS (row-major, padded)
        #pragma unroll
        for (int ct = 0; ct < 4; ++ct)
            #pragma unroll
            for (int i = 0; i < 8; ++i)
                pw[(rbase + i) * PROW + ct * 16 + li] = __expf(sc[ct][i] - m[i]);

        __builtin_amdgcn_wave_barrier();   // keep ds_store -> ds_load order

        // r = p(16x64) @ E(64x64): 16 K-slices of WMMA f32 16x16x4 per tile
        v8f acc[4] = {v8f{}, v8f{}, v8f{}, v8f{}};
        #pragma unroll
        for (int ks = 0; ks < 16; ++ks) {
            // A 16x4 fragment: lanes0-15 hold K=4ks..4ks+1, lanes16-31 K+2..3
            v2f a = *(const v2f*)&pw[li * PROW + 4 * ks + 2 * hi];
            #pragma unroll
            for (int ct = 0; ct < 4; ++ct) {
                v2f b = *(const v2f*)&eFrag[((ks * 4 + ct) * 32 + lane) * 2];
                acc[ct] = __builtin_amdgcn_wmma_f32_16x16x4_f32(
                    false, a, false, b, (short)0, acc[ct], false, false);
            }
        }

        // score' = em + m + log(r), masked select
        #pragma unroll
        for (int ct = 0; ct < 4; ++ct) {
            int col = ct * 16 + li;
            #pragma unroll
            for (int i = 0; i < 8; ++i) {
                int row = b0 + rbase + i;
                float e  = em[(long long)row * TK + (long long)t * K_ST + col];
                float nx = e + m[i] + __logf(acc[ct][i]);
                sc[ct][i] = mask[row * T_LEN + t] ? nx : sc[ct][i];
            }
        }
    }

    // ---- logZ = logsumexp(score + end) per row ----
    v8f m = {-3.4e38f, -3.4e38f, -3.4e38f, -3.4e38f,
             -3.4e38f, -3.4e38f, -3.4e38f, -3.4e38f};
    v8f val[4];
    #pragma unroll
    for (int ct = 0; ct < 4; ++ct) {
        float ev = sEnd[ct * 16 + li];
        #pragma unroll
        for (int i = 0; i < 8; ++i) {
            val[ct][i] = sc[ct][i] + ev;
            m[i] = fmaxf(m[i], val[ct][i]);
        }
    }
    #pragma unroll
    for (int off = 8; off >= 1; off >>= 1)
        #pragma unroll
        for (int i = 0; i < 8; ++i)
            m[i] = fmaxf(m[i], __shfl_xor(m[i], off));

    v8f s = {};
    #pragma unroll
    for (int ct = 0; ct < 4; ++ct)
        #pragma unroll
        for (int i = 0; i < 8; ++i)
            s[i] += __expf(val[ct][i] - m[i]);
    #pragma unroll
    for (int off = 8; off >= 1; off >>= 1)
        #pragma unroll
        for (int i = 0; i < 8; ++i)
            s[i] += __shfl_xor(s[i], off);

    if (li == 0) {  // lane 0 writes rows 0-7, lane 16 writes rows 8-15
        #pragma unroll
        for (int i = 0; i < 8; ++i)
            logZ[b0 + rbase + i] = m[i] + __logf(s[i]);
    }
}

// ---------------------------------------------------------------------------
// Gold score + final mean(logZ - gold).  One block, thread b = batch b.
// ---------------------------------------------------------------------------
__global__ __launch_bounds__(512) void crf_gold_reduce_kernel(
    const float* __restrict__ em,
    const long long* __restrict__ tags,      // int64 (B,T)
    const unsigned char* __restrict__ mask,  // (B,T)
    const float* __restrict__ trans,
    const float* __restrict__ startT,
    const float* __restrict__ endT,
    const float* __restrict__ logZ,
    float* __restrict__ out)
{
    const int b = threadIdx.x;
    const long long base = (long long)b * TK;
    const long long* tg = tags + (long long)b * T_LEN;
    const unsigned char* mk = mask + (long long)b * T_LEN;

    int prev = (int)tg[0];
    float score = startT[prev];
    int cnt = 0;
    for (int t = 0; t < T_LEN; ++t) {
        int tgt = (int)tg[t];
        float mf = mk[t] ? 1.0f : 0.0f;
        score += em[base + (long long)t * K_ST + tgt] * mf;
        if (t > 0) score += trans[prev * K_ST + tgt] * mf;
        prev = tgt;
        cnt += mk[t] ? 1 : 0;
    }
    score += endT[(int)tg[cnt - 1]];

    __shared__ float red[512];
    red[b] = logZ[b] - score;
    __syncthreads();
    for (int stp = 256; stp > 0; stp >>= 1) {
        if (b < stp) red[b] += red[b + stp];
        __syncthreads();
    }
    if (b == 0) out[0] = red[0] / (float)B_TOT;
}

// ---------------------------------------------------------------------------
extern "C" void kernel_launch(void* const* d_in, const int* in_sizes, int n_in,
                              void* d_out, int out_size, void* d_ws, size_t ws_size,
                              hipStream_t stream)
{
    (void)in_sizes; (void)n_in; (void)out_size; (void)ws_size;
    const float*         em     = (const float*)d_in[0];
    const long long*     tags   = (const long long*)d_in[1];
    const unsigned char* mask   = (const unsigned char*)d_in[2];
    const float*         trans  = (const float*)d_in[3];
    const float*         startT = (const float*)d_in[4];
    const float*         endT   = (const float*)d_in[5];
    float* out  = (float*)d_out;
    float* logZ = (float*)d_ws;   // 512 floats

    const int blocks = B_TOT / (WAVES_PER_BLOCK * ROWS_PER_WAVE); // 8
    crf_forward_kernel<<<blocks, WAVES_PER_BLOCK * 32, 0, stream>>>(
        em, mask, trans, startT, endT, logZ);
    crf_gold_reduce_kernel<<<1, B_TOT, 0, stream>>>(
        em, tags, mask, trans, startT, endT, logZ, out);
}